// CRvNN_transparent_32341103738881
// MI455X (gfx1250) — compile-verified
//
#include <hip/hip_runtime.h>
#include <hip/hip_bf16.h>
#include <math.h>

typedef float v2f __attribute__((ext_vector_type(2)));
typedef float v8f __attribute__((ext_vector_type(8)));

#define SS 32      // padded sequence length
#define S0 30      // raw sequence length
#define DD 256     // model dim
#define CHN 512    // cell hidden dim
#define NTHREADS 256
#define NWAVES 8

// ---- LDS layout (floats) -------------------------------------------------
// seq   [32*256]  @ 0
// L1    [32*256]  @ 8192    (left child, persists through step)
// INT   [32*512]  @ 16384   (L2 @ 16384, R1 @ 24576 during window phase)
// R2    [32*256]  @ 32768
// Hh    [32*256]  @ 40960   (conv h, then blended contents / trans)
// Lp    [32*32]   @ 49152
// Rp    [32*32]   @ 50176
// small [256]     @ 51200
#define OFF_SEQ   0
#define OFF_L1    8192
#define OFF_INT   16384
#define OFF_L2    16384
#define OFF_R1    24576
#define OFF_R2    32768
#define OFF_H     40960
#define OFF_LP    49152
#define OFF_RP    50176
#define OFF_SMALL 51200
#define SMEM_FLOATS 51456

__device__ __forceinline__ float gelu_exact(float x) {
  return 0.5f * x * (1.0f + erff(x * 0.70710678118654752f));
}
__device__ __forceinline__ float sigm(float x) { return 1.0f / (1.0f + expf(-x)); }

// One 16x16 output tile accumulation with V_WMMA_F32_16X16X4_F32.
// A: (16 x K) row-major (row stride lda), B: (K x 16) row-major (row stride ldb).
// f32 A-frag layout (ISA 7.12.2): lanes 0-15 M=0-15 with K=0,1 in v[0],v[1];
// lanes 16-31 hold K=2,3. B mirrors with N on the lane index.
__device__ __forceinline__ v8f wmma_mm(const float* __restrict__ A, int lda,
                                       const float* __restrict__ B, int ldb,
                                       int K, v8f acc, int lane) {
  const int half = lane >> 4;   // 0: K pair {0,1}, 1: K pair {2,3}
  const int idx  = lane & 15;   // A row / B col
  const float* ap = A + (size_t)idx * lda + 2 * half;
  const float* bp = B + (size_t)(2 * half) * ldb + idx;
  for (int k = 0; k < K; k += 4) {
    v2f a, b;
    a.x = ap[k];
    a.y = ap[k + 1];
    const float* bk = bp + (size_t)k * ldb;
    b.x = bk[0];
    b.y = bk[ldb];
    acc = __builtin_amdgcn_wmma_f32_16x16x4_f32(false, a, false, b, (short)0, acc,
                                                false, false);
  }
  return acc;
}

// Generic: C(32 x N, LDS, stride ldc) = A(32 x K, stride lda) @ B(K x N, stride ldb) + bias
__device__ __forceinline__ void mm32(const float* A, int lda, const float* B, int ldb,
                                     int K, int N, float* C, int ldc,
                                     const float* bias, int wave, int lane) {
  const int ntiles = 2 * (N >> 4);
  for (int t = wave; t < ntiles; t += NWAVES) {
    const int m0 = (t & 1) << 4;
    const int n0 = (t >> 1) << 4;
    v8f acc = {};
    acc = wmma_mm(A + (size_t)m0 * lda, lda, B + n0, ldb, K, acc, lane);
    const int col  = n0 + (lane & 15);
    const int half = lane >> 4;
    const float bs = bias ? bias[col] : 0.0f;
    float* cp = C + (size_t)(m0 + 8 * half) * ldc + col;
#pragma unroll
    for (int r = 0; r < 8; ++r) cp[(size_t)r * ldc] = acc[r] + bs;
  }
}

// Row-wise layernorm over DD features, one row per (wave, q); optional row scale.
__device__ __forceinline__ void layernorm_rows(float* buf, int ld, const float* g,
                                               const float* bta, const float* rowscale,
                                               int wave, int lane) {
  for (int q = 0; q < 4; ++q) {
    const int r = wave * 4 + q;
    float* p = buf + (size_t)r * ld;
    float s = 0.f, s2 = 0.f;
    for (int d = lane; d < DD; d += 32) { float v = p[d]; s += v; s2 += v * v; }
    for (int off = 16; off; off >>= 1) {
      s  += __shfl_xor(s, off, 32);
      s2 += __shfl_xor(s2, off, 32);
    }
    const float mu  = s * (1.0f / DD);
    const float var = s2 * (1.0f / DD) - mu * mu;
    const float rs  = rsqrtf(var + 1e-5f);
    const float sc  = rowscale ? rowscale[r] : 1.0f;
    for (int d = lane; d < DD; d += 32)
      p[d] = ((p[d] - mu) * rs * g[d] + bta[d]) * sc;
  }
}

__global__ __launch_bounds__(NTHREADS) void crvnn_kernel(
    const float* __restrict__ x, const float* __restrict__ msk,
    const float* __restrict__ START, const float* __restrict__ END,
    const float* __restrict__ Winit, const float* __restrict__ binit,
    const float* __restrict__ lng, const float* __restrict__ lnb,
    const float* __restrict__ Wconv, const float* __restrict__ bconv,
    const float* __restrict__ Wsc, const float* __restrict__ bsc,
    const float* __restrict__ Wc1, const float* __restrict__ bc1,
    const float* __restrict__ Wc2, const float* __restrict__ bc2,
    float* __restrict__ out) {
  extern __shared__ float sm[];
  float* seq = sm + OFF_SEQ;
  float* L1  = sm + OFF_L1;
  float* INTm = sm + OFF_INT;   // 32 x 512, stride CHN
  float* L2  = sm + OFF_L2;
  float* R1  = sm + OFF_R1;
  float* R2  = sm + OFF_R2;
  float* Hh  = sm + OFF_H;
  float* Lp  = sm + OFF_LP;
  float* Rp  = sm + OFF_RP;
  float* im_  = sm + OFF_SMALL;       // 32
  float* act  = sm + OFF_SMALL + 32;  // 32
  float* selp = sm + OFF_SMALL + 64;  // 32
  float* ltm  = sm + OFF_SMALL + 96;  // 32 last_token_mask
  float* tpA  = sm + OFF_SMALL + 128; // 32 (ts, then tp)
  float* em_  = sm + OFF_SMALL + 160; // 32 end_mask
  float* avv  = sm + OFF_SMALL + 192; // 32 active*im
  float* fl   = sm + OFF_SMALL + 224; // flags

  const int n = blockIdx.x;
  const int tid = threadIdx.x;
  const int wave = tid >> 5;
  const int lane = tid & 31;

  // ---- masks ----
  if (tid < SS) {
    const int j = tid;
    auto imbase = [&](int q) -> float {
      return (q == 0) ? 1.0f : ((q <= S0) ? msk[(size_t)n * S0 + (q - 1)] : 0.0f);
    };
    const float ine = (j < SS - 1) ? imbase(j) : 0.0f;
    const float iye = (j == 0) ? 1.0f : imbase(j - 1);
    const float em  = iye - ine;
    float em1 = 0.0f;
    if (j < SS - 1) {
      const int q = j + 1;
      const float ine1 = (q < SS - 1) ? imbase(q) : 0.0f;
      const float iye1 = imbase(q - 1);
      em1 = iye1 - ine1;
    }
    im_[j] = iye;
    em_[j] = em;
    ltm[j] = em1;
    selp[j] = ((j == 0) ? 0.0f : iye) * ine * (1.0f - em1);
    act[j] = iye;  // active = 1 * im
  }
  __syncthreads();

  // ---- raw seq into L2 region as staging X, with END blend ----
  for (int idx = tid; idx < SS * DD; idx += NTHREADS) {
    const int j = idx >> 8, d = idx & 255;
    float v;
    if (j == 0)        v = START[d];
    else if (j <= S0)  v = x[(size_t)n * S0 * DD + (size_t)(j - 1) * DD + d];
    else               v = 0.0f;
    const float em = em_[j];
    L2[idx] = em * END[d] + (1.0f - em) * v;
  }
  __syncthreads();

  // seq = layernorm(X @ Winit + binit) * im
  mm32(L2, DD, Winit, DD, DD, DD, seq, DD, binit, wave, lane);
  __syncthreads();
  layernorm_rows(seq, DD, lng, lnb, im_, wave, lane);
  __syncthreads();

  // ---- scan ----
  for (int it = 0; it < SS - 2; ++it) {
    // stop condition (equivalent to the u-gate freeze)
    if (tid == 0) {
      float rem = 0.0f;
      for (int j = 0; j < SS; ++j) rem += act[j] * selp[j];
      fl[0] = rem;
    }
    __syncthreads();
    if (fl[0] <= 0.01f) break;

    if (tid < SS) avv[tid] = act[tid] * im_[tid];
    __syncthreads();

    // neighbor probs: one row per lane (serial 32-wide cumsum, trivial cost)
    if (tid < SS) {
      const int t = tid;
      float cs = 0.0f;
      for (int j = 0; j < SS; ++j) {           // rightward channel -> Rp
        const float r   = (j > t) ? avv[j] : 0.0f;
        const float rem = (j == 0) ? 1.0f : fmaxf(1.0f - cs, 0.0f);
        cs += r;
        const float val = (cs > 1.0f) ? rem : r;
        Rp[t * SS + j] = val * im_[j];
      }
      cs = 0.0f;
      for (int j = 0; j < SS; ++j) {           // flipped channel -> Lp (double flip)
        const float a0  = avv[SS - 1 - j];
        const float m0  = im_[SS - 1 - j];
        const float r   = (j > t) ? a0 : 0.0f;
        const float rem = (j == 0) ? 1.0f : fmaxf(1.0f - cs, 0.0f);
        cs += r;
        const float val = (cs > 1.0f) ? rem : r;
        Lp[(SS - 1 - t) * SS + (SS - 1 - j)] = val * m0;
      }
    }
    __syncthreads();

    // window tensors
    mm32(Lp, SS, seq, DD, SS, DD, L1, DD, nullptr, wave, lane);   // L1 = Lp @ seq
    __syncthreads();
    mm32(Lp, SS, L1, DD, SS, DD, L2, DD, nullptr, wave, lane);    // L2 = Lp @ L1
    mm32(Rp, SS, seq, DD, SS, DD, R1, DD, nullptr, wave, lane);   // R1 = Rp @ seq
    __syncthreads();
    mm32(Rp, SS, R1, DD, SS, DD, R2, DD, nullptr, wave, lane);    // R2 = Rp @ R1
    __syncthreads();

    // h = gelu(win @ W_conv + b_conv), win = [L2,L1,seq,R1,R2]
    for (int t = wave; t < 32; t += NWAVES) {
      const int m0 = (t & 1) << 4, n0 = (t >> 1) << 4;
      v8f acc = {};
      acc = wmma_mm(L2  + (size_t)m0 * DD, DD, Wconv + 0 * DD * DD + n0, DD, DD, acc, lane);
      acc = wmma_mm(L1  + (size_t)m0 * DD, DD, Wconv + 1 * DD * DD + n0, DD, DD, acc, lane);
      acc = wmma_mm(seq + (size_t)m0 * DD, DD, Wconv + 2 * DD * DD + n0, DD, DD, acc, lane);
      acc = wmma_mm(R1  + (size_t)m0 * DD, DD, Wconv + 3 * DD * DD + n0, DD, DD, acc, lane);
      acc = wmma_mm(R2  + (size_t)m0 * DD, DD, Wconv + 4 * DD * DD + n0, DD, DD, acc, lane);
      const int col = n0 + (lane & 15);
      const int half = lane >> 4;
      const float bs = bconv[col];
      float* cp = Hh + (size_t)(m0 + 8 * half) * DD + col;
#pragma unroll
      for (int r = 0; r < 8; ++r) cp[(size_t)r * DD] = gelu_exact(acc[r] + bs);
    }
    __syncthreads();

    // transition scores -> tp (per-batch stabilized 2-way softmax)
    for (int q = 0; q < 4; ++q) {
      const int r = wave * 4 + q;
      float s = 0.0f;
      for (int d = lane; d < DD; d += 32) s += Hh[(size_t)r * DD + d] * Wsc[d];
      for (int off = 16; off; off >>= 1) s += __shfl_xor(s, off, 32);
      if (lane == 0) tpA[r] = s + bsc[0];
    }
    __syncthreads();
    if (tid == 0) {
      float g = 0.0f;
      for (int j = 0; j < SS; ++j) g = fmaxf(g, tpA[j]);
      const float e0 = expf(-g);
      for (int j = 0; j < SS; ++j) {
        const float e1 = expf(tpA[j] - g);
        tpA[j] = (e1 / (e1 + e0 + 1e-8f)) * selp[j];
      }
    }
    __syncthreads();

    // inter = gelu([L1, seq] @ W_c1 + b_c1)  (32 x 512)
    for (int t = wave; t < 64; t += NWAVES) {
      const int m0 = (t & 1) << 4, n0 = (t >> 1) << 4;
      v8f acc = {};
      acc = wmma_mm(L1  + (size_t)m0 * DD, DD, Wc1 + n0,             CHN, DD, acc, lane);
      acc = wmma_mm(seq + (size_t)m0 * DD, DD, Wc1 + 256 * CHN + n0, CHN, DD, acc, lane);
      const int col = n0 + (lane & 15);
      const int half = lane >> 4;
      const float bs = bc1[col];
      float* cp = INTm + (size_t)(m0 + 8 * half) * CHN + col;
#pragma unroll
      for (int r = 0; r < 8; ++r) cp[(size_t)r * CHN] = gelu_exact(acc[r] + bs);
    }
    __syncthreads();

    // contents = inter @ W_c2 + b_c2 (4 chunks), fused gated blend into Hh:
    // Hh = sig(C0)*L1 + sig(C1)*seq + sig(C2)*C3 ; fixed tile ownership, no
    // barriers needed between passes (each wave only touches its own tiles).
    for (int c = 3; c >= 0; --c) {
      for (int t = wave; t < 32; t += NWAVES) {
        const int m0 = (t & 1) << 4, n0 = (t >> 1) << 4;
        v8f acc = {};
        acc = wmma_mm(INTm + (size_t)m0 * CHN, CHN, Wc2 + c * DD + n0, 4 * DD, CHN, acc, lane);
        const int col = n0 + (lane & 15);
        const int half = lane >> 4;
        const float bs = bc2[c * DD + col];
        const size_t base = (size_t)(m0 + 8 * half) * DD + col;
#pragma unroll
        for (int r = 0; r < 8; ++r) {
          const float v = acc[r] + bs;
          const size_t o = base + (size_t)r * DD;
          if (c == 3)      Hh[o] = v;
          else if (c == 2) Hh[o] = sigm(v) * Hh[o];
          else if (c == 1) Hh[o] += sigm(v) * seq[o];
          else             Hh[o] += sigm(v) * L1[o];
        }
      }
    }
    __syncthreads();

    // trans = layernorm(blend)
    layernorm_rows(Hh, DD, lng, lnb, nullptr, wave, lane);
    __syncthreads();

    // seq <- (tp*trans + (1-tp)*seq) * im ; active <- max(active - Lp^T tp, 0)
    if (tid < SS) {
      const int j = tid;
      float cons = 0.0f;
      for (int i = 0; i < SS; ++i) cons += Lp[i * SS + j] * tpA[i];
      act[j] = fmaxf(act[j] - cons, 0.0f);
    }
    for (int idx = tid; idx < SS * DD; idx += NTHREADS) {
      const int j = idx >> 8;
      const float t_ = tpA[j];
      seq[idx] = (t_ * Hh[idx] + (1.0f - t_) * seq[idx]) * im_[j];
    }
    __syncthreads();
  }

  // output: sum over last_token_mask
  __syncthreads();
  if (tid < DD) {
    float s = 0.0f;
    for (int j = 0; j < SS; ++j) s += ltm[j] * seq[(size_t)j * DD + tid];
    out[(size_t)n * DD + tid] = s;
  }
}

extern "C" void kernel_launch(void* const* d_in, const int* in_sizes, int n_in,
                              void* d_out, int out_size, void* d_ws, size_t ws_size,
                              hipStream_t stream) {
  const float* x     = (const float*)d_in[0];
  const float* msk   = (const float*)d_in[1];
  const float* START = (const float*)d_in[2];
  const float* END   = (const float*)d_in[3];
  const float* Winit = (const float*)d_in[4];
  const float* binit = (const float*)d_in[5];
  const float* lng   = (const float*)d_in[6];
  const float* lnb   = (const float*)d_in[7];
  const float* Wconv = (const float*)d_in[8];
  const float* bconv = (const float*)d_in[9];
  const float* Wsc   = (const float*)d_in[10];
  const float* bsc   = (const float*)d_in[11];
  const float* Wc1   = (const float*)d_in[12];
  const float* bc1   = (const float*)d_in[13];
  const float* Wc2   = (const float*)d_in[14];
  const float* bc2   = (const float*)d_in[15];

  const int N = in_sizes[0] / (S0 * DD);
  const size_t smem = (size_t)SMEM_FLOATS * sizeof(float);
  crvnn_kernel<<<dim3(N), dim3(NTHREADS), smem, stream>>>(
      x, msk, START, END, Winit, binit, lng, lnb, Wconv, bconv, Wsc, bsc,
      Wc1, bc1, Wc2, bc2, (float*)d_out);
}